// RWKV_Time_Mixing_34815004901973
// MI455X (gfx1250) — compile-verified
//
#include <hip/hip_runtime.h>
#include <hip/hip_bf16.h>

#define C_DIM 4096
#define KSPLIT 8
#define KCHUNK (C_DIM / KSPLIT)          // 512
#define WAVES_PER_BLOCK 8
#define ROWS_PER_WAVE 16
#define ROWS_PER_BLOCK (WAVES_PER_BLOCK * ROWS_PER_WAVE)  // 128
#define ROW_BLOCKS (C_DIM / ROWS_PER_BLOCK)               // 32
#define STATE_ROWS 160

typedef float v2f __attribute__((ext_vector_type(2)));
typedef float v8f __attribute__((ext_vector_type(8)));

// ---------------------------------------------------------------------------
// K1: token-shift mixing  xk/xv/xr = x*m + x_prev*(1-m)
// ---------------------------------------------------------------------------
__global__ __launch_bounds__(256) void prep_kernel(
    const float* __restrict__ x, const float* __restrict__ state,
    const int* __restrict__ layer_num,
    const float* __restrict__ tmk, const float* __restrict__ tmv,
    const float* __restrict__ tmr, float* __restrict__ ws_x) {
  int c = blockIdx.x * blockDim.x + threadIdx.x;
  int i = *layer_num;
  const float* xp = state + (size_t)(5 * i + 1) * C_DIM;
  float xc = x[c], xpc = xp[c];
  float mk = tmk[c], mv = tmv[c], mr = tmr[c];
  ws_x[c]             = xc * mk + xpc * (1.0f - mk);
  ws_x[C_DIM + c]     = xc * mv + xpc * (1.0f - mv);
  ws_x[2 * C_DIM + c] = xc * mr + xpc * (1.0f - mr);
}

// ---------------------------------------------------------------------------
// K2/K5: WMMA matvec partials.  Each wave computes 16 output rows over one
// K-chunk using V_WMMA_F32_16X16X4_F32.  A = 16x4 weight tile (lane L holds
// row L&15, K-offset 2*(L>=16)), B = x[k] broadcast across all 16 columns.
// D columns are identical; lanes with (lane&15)==0 hold rows (per C/D layout:
// lane 0 -> M=0..7, lane 16 -> M=8..15).  Weights are read exactly once
// (K-split gives each row-chunk a single owner wave) -> non-temporal loads
// so the 256 MB stream doesn't rinse through L2.
// ---------------------------------------------------------------------------
__global__ __launch_bounds__(256) void matvec_wmma(
    const float* __restrict__ w0, const float* __restrict__ w1,
    const float* __restrict__ w2, const float* __restrict__ xbase,
    float* __restrict__ partial) {
  __shared__ float xs[KCHUNK];

  const int mat = blockIdx.z;
  const float* W = (mat == 0) ? w0 : ((mat == 1) ? w1 : w2);
  const float* xvec = xbase + (size_t)mat * C_DIM;
  const int k0 = blockIdx.y * KCHUNK;

  // stage the x chunk into LDS (coalesced)
  for (int t = threadIdx.x; t < KCHUNK; t += blockDim.x)
    xs[t] = xvec[k0 + t];
  __syncthreads();

  const int wave = threadIdx.x >> 5;   // wave32
  const int lane = threadIdx.x & 31;
  const int hi   = lane >> 4;          // which K-half of the 16x4 tile
  const int m    = (blockIdx.x * WAVES_PER_BLOCK + wave) * ROWS_PER_WAVE +
                   (lane & 15);

  const float* wrow = W + (size_t)m * C_DIM + k0 + 2 * hi;  // 8B aligned
  const float* xsrc = xs + 2 * hi;

  v8f acc = {};
#pragma unroll 8
  for (int k = 0; k < KCHUNK; k += 4) {
    // streaming, read-once weights: non-temporal b64 load
    v2f a = __builtin_nontemporal_load((const v2f*)(wrow + k));
    v2f b = *(const v2f*)(xsrc + k);   // LDS broadcast
    acc = __builtin_amdgcn_wmma_f32_16x16x4_f32(
        /*neg_a=*/false, a, /*neg_b=*/false, b,
        /*c_mod=*/(short)0, acc, /*reuse_a=*/false, /*reuse_b=*/false);
  }

  if ((lane & 15) == 0) {
    int rowbase = (blockIdx.x * WAVES_PER_BLOCK + wave) * ROWS_PER_WAVE + hi * 8;
    float* dst = partial + ((size_t)mat * KSPLIT + blockIdx.y) * C_DIM + rowbase;
#pragma unroll
    for (int v = 0; v < 8; ++v) dst[v] = acc[v];
  }
}

// ---------------------------------------------------------------------------
// K4: reduce K/V/R partials (fixed order -> deterministic), sigmoid, WKV with
// log-sum-exp max trick, write updated state rows into d_out, r*wkv into ws.
// ---------------------------------------------------------------------------
__global__ __launch_bounds__(256) void wkv_kernel(
    const float* __restrict__ state, const int* __restrict__ layer_num,
    const float* __restrict__ time_first, const float* __restrict__ time_decay,
    const float* __restrict__ partial, float* __restrict__ ws_rwkv,
    float* __restrict__ out_state) {
  int c = blockIdx.x * blockDim.x + threadIdx.x;
  int i = *layer_num;

  float kk = 0.f, vv = 0.f, rr = 0.f;
#pragma unroll
  for (int s = 0; s < KSPLIT; ++s) {
    kk += partial[(0 * KSPLIT + s) * C_DIM + c];
    vv += partial[(1 * KSPLIT + s) * C_DIM + c];
    rr += partial[(2 * KSPLIT + s) * C_DIM + c];
  }

  float r  = 1.0f / (1.0f + __expf(-rr));
  float aa = state[(size_t)(5 * i + 2) * C_DIM + c];
  float bb = state[(size_t)(5 * i + 3) * C_DIM + c];
  float pp = state[(size_t)(5 * i + 4) * C_DIM + c];

  float ww = time_first[c] + kk;
  float p  = fmaxf(pp, ww);
  float e1 = __expf(pp - p);
  float e2 = __expf(ww - p);
  float a  = e1 * aa + e2 * vv;
  float b  = e1 * bb + e2;

  float ww2 = pp + time_decay[c];
  float p2  = fmaxf(ww2, kk);
  float e1b = __expf(ww2 - p2);
  float e2b = __expf(kk - p2);

  out_state[(size_t)(5 * i + 2) * C_DIM + c] = e1b * aa + e2b * vv;
  out_state[(size_t)(5 * i + 3) * C_DIM + c] = e1b * bb + e2b;
  out_state[(size_t)(5 * i + 4) * C_DIM + c] = p2;

  ws_rwkv[c] = r * (a / b);
}

// ---------------------------------------------------------------------------
// K6: final reduce of O-matvec partials into the output vector
// ---------------------------------------------------------------------------
__global__ __launch_bounds__(256) void reduce_out(
    const float* __restrict__ partialO, float* __restrict__ outv) {
  int c = blockIdx.x * blockDim.x + threadIdx.x;
  float s = 0.f;
#pragma unroll
  for (int j = 0; j < KSPLIT; ++j) s += partialO[j * C_DIM + c];
  outv[c] = s;
}

// ---------------------------------------------------------------------------
extern "C" void kernel_launch(void* const* d_in, const int* in_sizes, int n_in,
                              void* d_out, int out_size, void* d_ws,
                              size_t ws_size, hipStream_t stream) {
  const float* x     = (const float*)d_in[0];
  const float* state = (const float*)d_in[1];
  const int*   lnum  = (const int*)d_in[2];
  const float* tmk   = (const float*)d_in[3];
  const float* tmv   = (const float*)d_in[4];
  const float* tmr   = (const float*)d_in[5];
  const float* tfirst= (const float*)d_in[6];
  const float* tdecay= (const float*)d_in[7];
  const float* kw    = (const float*)d_in[8];
  const float* vw    = (const float*)d_in[9];
  const float* rw    = (const float*)d_in[10];
  const float* ow    = (const float*)d_in[11];

  float* out_state = (float*)d_out;                          // 160*4096
  float* out_vec   = out_state + (size_t)STATE_ROWS * C_DIM; // 4096

  // workspace layout (floats)
  float* ws_x    = (float*)d_ws;                 // 3*C   : xk,xv,xr
  float* ws_pKVR = ws_x + 3 * C_DIM;             // 3*KSPLIT*C
  float* ws_rwkv = ws_pKVR + 3 * KSPLIT * C_DIM; // C
  float* ws_pO   = ws_rwkv + C_DIM;              // KSPLIT*C

  dim3 blk(256);

  // 1) token-shift mix
  prep_kernel<<<dim3(C_DIM / 256), blk, 0, stream>>>(x, state, lnum, tmk, tmv,
                                                     tmr, ws_x);

  // 2) kk / vv / rr partials via WMMA
  matvec_wmma<<<dim3(ROW_BLOCKS, KSPLIT, 3), blk, 0, stream>>>(
      kw, vw, rw, ws_x, ws_pKVR);

  // 3) copy full input state to output (rows 5i+2..4 overwritten next)
  hipMemcpyAsync(out_state, state, (size_t)STATE_ROWS * C_DIM * sizeof(float),
                 hipMemcpyDeviceToDevice, stream);

  // 4) WKV elementwise + state row updates + r*wkv
  wkv_kernel<<<dim3(C_DIM / 256), blk, 0, stream>>>(
      state, lnum, tfirst, tdecay, ws_pKVR, ws_rwkv, out_state);

  // 5) ow @ (r*wkv) partials via WMMA (single matrix -> grid.z = 1)
  matvec_wmma<<<dim3(ROW_BLOCKS, KSPLIT, 1), blk, 0, stream>>>(
      ow, ow, ow, ws_rwkv, ws_pO);

  // 6) final reduction into output vector
  reduce_out<<<dim3(C_DIM / 256), blk, 0, stream>>>(ws_pO, out_vec);
}